// MultiHeadAttention_23218593202679
// MI455X (gfx1250) — compile-verified
//
#include <hip/hip_runtime.h>

// Problem constants (B,S,D,H from the reference)
#define B_    2
#define S_    2048
#define D_    1024
#define H_    16
#define DH_   64
#define NEGV  (-10000.0f)
#define LNEPS 1e-12f
#define SCL   0.03125f    // 1/sqrt(1024)
#define PAD_  2052        // LDS row stride (floats). Multiple of 4 -> rows are
                          // 16B aligned for async b128; 2052%64==4 -> stage-4
                          // strided row reads hit distinct banks.

typedef __attribute__((ext_vector_type(2))) float v2f;
typedef __attribute__((ext_vector_type(8))) float v8f;

// Pure-fp32 WMMA: D(16x16,f32) = A(16x4,f32) * B(4x16,f32) + C
__device__ __forceinline__ v8f wmma_f32(v2f a, v2f b, v8f c) {
  return __builtin_amdgcn_wmma_f32_16x16x4_f32(false, a, false, b, (short)0, c,
                                               false, false);
}

// ---------------------------------------------------------------------------
// Dense projection GEMM: out[M,N] = A[M,K] @ W[N,K]^T + bias (+ residual)
// M = B*S = 4096, N = K = D = 1024. Block = 8 waves -> 32(M) x 64(N) tile.
// Wq/Wo (4 MB) and A (16 MB) are resident in the 192 MB L2.
// ---------------------------------------------------------------------------
template <bool RES>
__global__ __launch_bounds__(256) void proj_kernel(
    const float* __restrict__ A, const float* __restrict__ W,
    const float* __restrict__ bias, const float* __restrict__ residual,
    float* __restrict__ out) {
  const int lane  = threadIdx.x & 31;
  const int wv    = threadIdx.x >> 5;              // 0..7
  const int m0    = blockIdx.y * 32 + (wv >> 2) * 16;
  const int n0    = blockIdx.x * 64 + (wv & 3) * 16;
  const int mr    = m0 + (lane & 15);              // A row for this lane
  const int nr    = n0 + (lane & 15);              // W row (output col)
  const int khalf = (lane >> 4) * 2;               // 0 or 2: K sub-offset

  v8f acc = {};
#pragma unroll 8
  for (int k0 = 0; k0 < D_; k0 += 4) {
    v2f a = *(const v2f*)(A + (size_t)mr * D_ + k0 + khalf);
    v2f b = *(const v2f*)(W + (size_t)nr * D_ + k0 + khalf);  // B[k][n]=W[n][k]
    acc = wmma_f32(a, b, acc);
  }

  const int rbase = m0 + ((lane >> 4) << 3);
  const int col   = n0 + (lane & 15);
  const float bb  = bias[col];
#pragma unroll
  for (int v = 0; v < 8; ++v) {
    const size_t idx = (size_t)(rbase + v) * D_ + col;
    if (RES)
      out[idx] = acc[v] + bb + residual[idx];
    else
      out[idx] = acc[v] + bb;
  }
}

// ---------------------------------------------------------------------------
// Fused attention core. One block per (bh, 32-query stripe):
//   stage 1: scores stripe (32x2048) -> LDS  (WMMA, K=64, masks applied)
//   stage 2: row softmax in LDS (half-wave per row)
//   stage 3: async LDS->global write of the stripe (GLOBAL_STORE_ASYNC_
//            FROM_LDS_B128, ASYNCcnt-tracked; the only mandatory HBM stream)
//   stage 4: ctx stripe (32x64) = probs(LDS) @ qh   (WMMA, K=2048)
// LDS: 32 * 2052 * 4 B = 256.5 KB (dynamic; CDNA5 WGP has 320 KB).
// ---------------------------------------------------------------------------
__global__ __launch_bounds__(256) void fused_attn_kernel(
    const float* __restrict__ q, const float* __restrict__ pm,
    float* __restrict__ P, float* __restrict__ ctx) {
  extern __shared__ float smem[];                  // [32][PAD_]

  const int bh    = blockIdx.y;                    // 0..31
  const int b     = bh >> 4, h = bh & 15;
  const int qbase = blockIdx.x * 32;               // query stripe start
  const int t     = threadIdx.x;
  const int lane  = t & 31;
  const int wv    = t >> 5;                        // 0..7
  const int khalf = (lane >> 4) * 2;

  const float* qh = q + (size_t)b * S_ * D_ + h * DH_;   // row stride D_

  // ---- stage 1: scores -> LDS -------------------------------------------
  {
    const int mrow0 = (wv >> 2) * 16;              // 0 or 16 (stripe-local)
    const int qrow  = qbase + mrow0 + (lane & 15);
    v2f afrag[16];                                 // A 16x64 fragment, reused
#pragma unroll
    for (int kk = 0; kk < 16; ++kk)
      afrag[kk] = *(const v2f*)(qh + (size_t)qrow * D_ + kk * 4 + khalf);

    for (int ct = 0; ct < 32; ++ct) {              // 32 column tiles / wave
      const int n0  = (wv & 3) * 512 + ct * 16;
      const int col = n0 + (lane & 15);
      const float* qcol = qh + (size_t)col * D_;   // B[k][n] = qh[n][k]

      // Batch all 16 B fragments into distinct registers so the scheduler
      // can issue them as one load clause and drain with staged loadcnt.
      v2f bfrag[16];
#pragma unroll
      for (int kk = 0; kk < 16; ++kk)
        bfrag[kk] = *(const v2f*)(qcol + kk * 4 + khalf);

      v8f acc = {};
#pragma unroll
      for (int kk = 0; kk < 16; ++kk)
        acc = wmma_f32(afrag[kk], bfrag[kk], acc);

      const float padd = (1.0f - pm[b * S_ + col]) * NEGV;
#pragma unroll
      for (int v = 0; v < 8; ++v) {
        const int lrow = mrow0 + ((lane >> 4) << 3) + v;   // 0..31
        float s = acc[v] * SCL + padd;
        if (col > qbase + lrow) s = NEGV;          // causal masked_fill (set)
        smem[lrow * PAD_ + col] = s;
      }
    }
  }
  __syncthreads();

  // ---- stage 2: softmax, one row per 16-lane group (2 rows each) --------
  {
    const int li = lane & 15;
#pragma unroll
    for (int rr = 0; rr < 2; ++rr) {
      float* prow = smem + ((t >> 4) + rr * 16) * PAD_;
      float m = -3.402823466e38f;
      for (int i = li; i < S_; i += 16) m = fmaxf(m, prow[i]);
#pragma unroll
      for (int off = 8; off > 0; off >>= 1) m = fmaxf(m, __shfl_xor(m, off, 32));
      float sum = 0.0f;
      for (int i = li; i < S_; i += 16) {
        float e = __expf(prow[i] - m);
        prow[i] = e;
        sum += e;
      }
#pragma unroll
      for (int off = 8; off > 0; off >>= 1) sum += __shfl_xor(sum, off, 32);
      const float inv = 1.0f / sum;
      for (int i = li; i < S_; i += 16) prow[i] *= inv;
    }
  }
  __syncthreads();  // DS writes drained before the async engine reads LDS

  // ---- stage 3: async LDS -> global stripe write ------------------------
  {
    float* Pbh = P + ((size_t)bh * S_ + qbase) * S_;
    const unsigned lds_base = (unsigned)(unsigned long long)smem;
    for (int it = 0; it < 64; ++it) {              // 256 thr * 16B * 64 = 256KB
      const int e4  = (t + it * 256) * 4;          // float index, mult. of 4
      const int row = e4 >> 11;                    // /2048
      const int col = e4 & (S_ - 1);
      const unsigned lds = lds_base + (unsigned)((row * PAD_ + col) * 4);
      const float* gp = Pbh + (size_t)row * S_ + col;
      asm volatile("global_store_async_from_lds_b128 %0, %1, off"
                   :: "v"(gp), "v"(lds)
                   : "memory");
    }
  }

  // ---- stage 4: ctx stripe = probs(LDS) @ qh ----------------------------
  {
    const int m0 = (wv >> 2) * 16;
    const int n0 = (wv & 3) * 16;
    const int mr = m0 + (lane & 15);
    const int nr = n0 + (lane & 15);

    // Software-pipeline the global B loads one iteration ahead of the WMMA.
    v2f bcur;
    bcur.x = qh[(size_t)khalf * D_ + nr];
    bcur.y = qh[(size_t)(khalf + 1) * D_ + nr];
    v8f acc = {};
#pragma unroll 4
    for (int k0 = 0; k0 < S_; k0 += 4) {
      v2f bnext;
      if (k0 + 4 < S_) {
        const int kn = k0 + 4 + khalf;
        bnext.x = qh[(size_t)kn * D_ + nr];
        bnext.y = qh[(size_t)(kn + 1) * D_ + nr];
      }
      v2f a = *(const v2f*)(smem + mr * PAD_ + k0 + khalf);  // LDS
      acc = wmma_f32(a, bcur, acc);
      bcur = bnext;
    }
    const int rbase = qbase + m0 + ((lane >> 4) << 3);
    const int col   = n0 + (lane & 15);
#pragma unroll
    for (int v = 0; v < 8; ++v)
      ctx[((size_t)b * S_ + rbase + v) * D_ + h * DH_ + col] = acc[v];
  }

  // Drain outstanding async LDS->global stores before wave end.
  asm volatile("s_wait_asynccnt 0x0" ::: "memory");
}

// ---------------------------------------------------------------------------
// In-place LayerNorm over D=1024; one 8-wave block per row.
// ---------------------------------------------------------------------------
__global__ __launch_bounds__(256) void ln_kernel(
    float* __restrict__ x, const float* __restrict__ gamma,
    const float* __restrict__ beta) {
  __shared__ float sred[8];
  float* xr = x + (size_t)blockIdx.x * D_;
  const int t = threadIdx.x, lane = t & 31, wv = t >> 5;

  float v[4];
  float s = 0.0f;
#pragma unroll
  for (int i = 0; i < 4; ++i) { v[i] = xr[t + i * 256]; s += v[i]; }
#pragma unroll
  for (int off = 16; off > 0; off >>= 1) s += __shfl_xor(s, off, 32);
  if (lane == 0) sred[wv] = s;
  __syncthreads();
  float t8 = (lane < 8) ? sred[lane] : 0.0f;
#pragma unroll
  for (int off = 4; off > 0; off >>= 1) t8 += __shfl_xor(t8, off, 32);
  const float mu = __shfl(t8, 0, 32) * (1.0f / D_);

  float ss = 0.0f;
#pragma unroll
  for (int i = 0; i < 4; ++i) { float d = v[i] - mu; ss += d * d; }
#pragma unroll
  for (int off = 16; off > 0; off >>= 1) ss += __shfl_xor(ss, off, 32);
  __syncthreads();
  if (lane == 0) sred[wv] = ss;
  __syncthreads();
  float s8 = (lane < 8) ? sred[lane] : 0.0f;
#pragma unroll
  for (int off = 4; off > 0; off >>= 1) s8 += __shfl_xor(s8, off, 32);
  const float rstd = rsqrtf(__shfl(s8, 0, 32) * (1.0f / D_) + LNEPS);

#pragma unroll
  for (int i = 0; i < 4; ++i) {
    const int c = t + i * 256;
    xr[c] = gamma[c] * (v[i] - mu) * rstd + beta[c];
  }
}

// ---------------------------------------------------------------------------
extern "C" void kernel_launch(void* const* d_in, const int* in_sizes, int n_in,
                              void* d_out, int out_size, void* d_ws,
                              size_t ws_size, hipStream_t stream) {
  const float* emb   = (const float*)d_in[0];
  const float* pm    = (const float*)d_in[1];
  const float* Wq    = (const float*)d_in[2];
  const float* bq    = (const float*)d_in[3];
  // d_in[4..7] = Wk,bk,Wv,bv: dead in the reference (K,V come from Q) — skipped.
  const float* Wo    = (const float*)d_in[8];
  const float* bo    = (const float*)d_in[9];
  const float* gamma = (const float*)d_in[10];
  const float* beta  = (const float*)d_in[11];

  float* out = (float*)d_out;
  float* ln  = out;                               // [B,S,D]
  float* P   = out + (size_t)B_ * S_ * D_;        // [B,H,S,S] (tuple order)

  float* q   = (float*)d_ws;                      // [B,S,D] 16 MB
  float* ctx = q + (size_t)B_ * S_ * D_;          // [B,S,D] 16 MB

  const dim3 blk(256);
  const size_t attn_lds = (size_t)32 * PAD_ * sizeof(float);  // 256.5 KB

  // 1) q = emb @ Wq^T + bq
  proj_kernel<false><<<dim3(D_ / 64, (B_ * S_) / 32), blk, 0, stream>>>(
      emb, Wq, bq, nullptr, q);
  // 2) fused scores + masks + softmax + async probs write + ctx
  fused_attn_kernel<<<dim3(S_ / 32, B_ * H_), blk, attn_lds, stream>>>(
      q, pm, P, ctx);
  // 3) x = ctx @ Wo^T + bo + emb  (written into ln region)
  proj_kernel<true><<<dim3(D_ / 64, (B_ * S_) / 32), blk, 0, stream>>>(
      ctx, Wo, bo, emb, ln);
  // 4) LayerNorm in place
  ln_kernel<<<dim3(B_ * S_), blk, 0, stream>>>(ln, gamma, beta);
}